// GNN123_80109730005711
// MI455X (gfx1250) — compile-verified
//
#include <hip/hip_runtime.h>

// ---------------- problem constants (match reference) ----------------
#define N1c 50000
#define N2c 100000
#define N3c 120000
#define E1c 1600000LL
#define E2c 1500000LL
#define E3c 1000000LL
#define A2c 200000LL
#define A3c 360000LL
#define Bc  256
#define NI2c 50
#define NI3c 100

typedef float v2f __attribute__((ext_vector_type(2)));
typedef float v4f __attribute__((ext_vector_type(4)));
typedef float v8f __attribute__((ext_vector_type(8)));

// ---------------- utility kernels ----------------
__global__ void k_zero(float* __restrict__ p, long long n) {
  long long n4 = n >> 2;
  long long i = (long long)blockIdx.x * blockDim.x + threadIdx.x;
  long long step = (long long)gridDim.x * blockDim.x;
  v4f z = {0.f, 0.f, 0.f, 0.f};
  for (long long j = i; j < n4; j += step) ((v4f*)p)[j] = z;
  long long rem = n & 3;
  if (i < rem) p[(n4 << 2) + i] = 0.f;
}

// 4 floats per thread: out[sidx[e]*os + oc + f..f+3] += src[(gidx?gidx[e]:e)*ss + f..f+3]
// fshift = log2(dim/4); dim is 32 or 64.
__global__ void k_scatter_add4(const float* __restrict__ src, int src_stride,
                               const int* __restrict__ gidx,
                               const int* __restrict__ sidx,
                               long long E, int fshift,
                               float* __restrict__ out, int out_stride, int out_col) {
  long long total = E << fshift;
  long long t = (long long)blockIdx.x * blockDim.x + threadIdx.x;
  long long step = (long long)gridDim.x * blockDim.x;
  for (; t < total; t += step) {
    long long e = t >> fshift;
    int f = (int)((t - (e << fshift)) << 2);
    int srow = gidx ? gidx[e] : (int)e;
    int drow = sidx[e];
    v4f v = *(const v4f*)(src + (long long)srow * src_stride + f);
    float* op = out + (long long)drow * out_stride + out_col + f;
    atomicAdd(op + 0, v.x);
    atomicAdd(op + 1, v.y);
    atomicAdd(op + 2, v.z);
    atomicAdd(op + 3, v.w);
  }
}

__global__ void k_count(const int* __restrict__ idx, long long n, float* __restrict__ cnt) {
  long long i = (long long)blockIdx.x * blockDim.x + threadIdx.x;
  long long step = (long long)gridDim.x * blockDim.x;
  for (; i < n; i += step) atomicAdd(&cnt[idx[i]], 1.0f);
}

// buf[i*stride + col + f] /= max(cnt[i], 1)
__global__ void k_div(float* __restrict__ buf, int stride, int col, int dim,
                      const float* __restrict__ cnt, int n) {
  long long total = (long long)n * dim;
  long long t = (long long)blockIdx.x * blockDim.x + threadIdx.x;
  long long step = (long long)gridDim.x * blockDim.x;
  for (; t < total; t += step) {
    long long i = t / dim;
    int f = (int)(t - i * dim);
    buf[i * stride + col + f] /= fmaxf(cnt[i], 1.0f);
  }
}

// ---------------- WMMA fp32 GEMM: out = act(A @ Wslice^T + C0 + bias) ----------------
// A element:  A[row*lda + k]; W element: W[col*ldw + k]  (B[k][col] = W^T slice).
// C0 optional additive [N, Cout] (may alias out). K is compile-time.
// One wave per 16x16 tile; block = 32 * (Cout/16) threads.
template <int K>
__global__ __launch_bounds__(128)
void k_wmma_gemm(const float* __restrict__ A, int lda,
                 const float* __restrict__ W, int ldw,
                 const float* __restrict__ C0,
                 const float* __restrict__ bias,
                 float* __restrict__ out,
                 int N, int Cout, int do_elu) {
  constexpr int KF = K & ~3;   // full 4-wide steps
  const int lane = threadIdx.x & 31;
  const int wave = threadIdx.x >> 5;
  const int rbase = blockIdx.x << 4;
  int arow = rbase + (lane & 15);
  if (arow >= N) arow = N - 1;               // clamp; stores predicated below
  const int col = (wave << 4) + (lane & 15);
  const int khi = (lane < 16) ? 0 : 2;       // A/B: lanes0-15 K={0,1}, lanes16-31 K={2,3}

  const float* ap = A + (long long)arow * lda + khi;  // 8B-aligned (lda, khi even)
  const float* wp = W + (long long)col * ldw + khi;

  v8f acc = {0.f, 0.f, 0.f, 0.f, 0.f, 0.f, 0.f, 0.f};
#pragma unroll 8
  for (int k0 = 0; k0 < KF; k0 += 4) {
    v2f a = *(const v2f*)(ap + k0);          // unconditional b64 loads
    v2f b = *(const v2f*)(wp + k0);
    acc = __builtin_amdgcn_wmma_f32_16x16x4_f32(false, a, false, b,
                                                (short)0, acc, false, false);
  }
  if constexpr ((K & 3) != 0) {              // one-time tail (K%4==2 for K=50)
    float a0 = 0.f, a1 = 0.f, b0 = 0.f, b1 = 0.f;
    if (KF + khi < K)     { a0 = ap[KF];     b0 = wp[KF]; }
    if (KF + khi + 1 < K) { a1 = ap[KF + 1]; b1 = wp[KF + 1]; }
    v2f a = {a0, a1}, b = {b0, b1};
    acc = __builtin_amdgcn_wmma_f32_16x16x4_f32(false, a, false, b,
                                                (short)0, acc, false, false);
  }

  const float bv = bias ? bias[col] : 0.f;
#pragma unroll
  for (int v = 0; v < 8; ++v) {
    const int grow = rbase + v + ((lane < 16) ? 0 : 8);  // D: VGPR v -> M=v (+8 hi lanes)
    if (grow < N) {
      long long oi = (long long)grow * Cout + col;
      float y = acc[v] + bv;
      if (C0) y += C0[oi];
      if (do_elu) y = (y > 0.f) ? y : (expf(y) - 1.f);
      out[oi] = y;
    }
  }
}

// ---------------- fc3 (32 -> 11) + log_softmax, one row per block ----------------
__global__ void k_fc3_lsm(const float* __restrict__ tin, const float* __restrict__ w,
                          const float* __restrict__ b, float* __restrict__ out) {
  const int row = blockIdx.x;
  const int t = threadIdx.x;  // 32 threads
  __shared__ float logits[11];
  __shared__ float smax, slse;
  if (t < 11) {
    float s = b[t];
    const float* r = tin + row * 32;
    const float* wr = w + t * 32;
#pragma unroll
    for (int k = 0; k < 32; ++k) s += r[k] * wr[k];
    logits[t] = s;
  }
  __syncthreads();
  if (t == 0) {
    float mx = logits[0];
    for (int i = 1; i < 11; ++i) mx = fmaxf(mx, logits[i]);
    float s = 0.f;
    for (int i = 0; i < 11; ++i) s += expf(logits[i] - mx);
    smax = mx;
    slse = logf(s);
  }
  __syncthreads();
  if (t < 11) out[row * 11 + t] = logits[t] - smax - slse;
}

// ---------------- host orchestration ----------------
static inline int gcap(long long blocks) {
  if (blocks < 1) blocks = 1;
  if (blocks > 262144) blocks = 262144;
  return (int)blocks;
}

extern "C" void kernel_launch(void* const* d_in, const int* in_sizes, int n_in,
                              void* d_out, int out_size, void* d_ws, size_t ws_size,
                              hipStream_t stream) {
  (void)in_sizes; (void)n_in; (void)out_size;
  const float* x    = (const float*)d_in[0];
  const int*   ei1  = (const int*)d_in[1];   // [2,E1]: row0=src, row1=dst
  const int*   ei2  = (const int*)d_in[2];
  const int*   ei3  = (const int*)d_in[3];
  const int*   ai2  = (const int*)d_in[4];   // [2,A2]: row0=N1 idx, row1=N2 idx
  const int*   ai3  = (const int*)d_in[5];
  const float* iso2 = (const float*)d_in[6]; // [N2,50]
  const float* iso3 = (const float*)d_in[7]; // [N3,100]
  const int*   bat1 = (const int*)d_in[8];
  const int*   bat2 = (const int*)d_in[9];
  const int*   bat3 = (const int*)d_in[10];
  const float *c_wr[7], *c_wl[7], *c_b[7];   // w_root, w_rel, bias per conv (dict order)
  for (int i = 0; i < 7; ++i) {
    c_wr[i] = (const float*)d_in[11 + 3 * i];
    c_wl[i] = (const float*)d_in[12 + 3 * i];
    c_b[i]  = (const float*)d_in[13 + 3 * i];
  }
  const float* fw1 = (const float*)d_in[32]; const float* fb1 = (const float*)d_in[33];
  const float* fw2 = (const float*)d_in[34]; const float* fb2 = (const float*)d_in[35];
  const float* fw3 = (const float*)d_in[36]; const float* fb3 = (const float*)d_in[37];

  // workspace layout (floats); every offset is a multiple of 4 (16B alignment)
  float* W = (float*)d_ws;
  const long long F_G  = 0;                       // [256,192] pooled graph features
  const long long F_CB = F_G  + 256LL * 192;      // [256] batch counts
  const long long F_CN = F_CB + 256;              // [max(N2,N3)] tuple counts
  const long long F_HA = F_CN + 120000;           // [N1,64] h ping
  const long long F_HB = F_HA + 3200000;          // [N1,64] h pong (final level-1 h)
  const long long F_XR = F_HB + 3200000;          // [maxN,64] projected features
  const long long F_AG = F_XR + 7680000;          // [maxN,64] edge-aggregated
  const long long F_P2 = F_AG + 7680000;          // [maxN,64] pooled / h2b / h3b
  const long long F_HC = F_P2 + 7680000;          // [maxN,64] h2a / h3a
  const long long F_END = F_HC + 7680000;         // ~149 MB total
  if (ws_size < (size_t)F_END * sizeof(float)) return;
  float *G = W + F_G, *CB = W + F_CB, *CN = W + F_CN, *HA = W + F_HA, *HB = W + F_HB,
        *XR = W + F_XR, *AG = W + F_AG, *P2 = W + F_P2, *HC = W + F_HC;

  auto zero = [&](float* p, long long n) {
    k_zero<<<gcap((n / 4 + 255) / 256), 256, 0, stream>>>(p, n);
  };
  auto scat = [&](const float* src, int ss, const int* g, const int* s, long long E,
                  int dim, float* o, int os, int oc) {
    int fshift = (dim == 64) ? 4 : 3;  // dim/4 = 16 or 8
    k_scatter_add4<<<gcap(((E << fshift) + 255) / 256), 256, 0, stream>>>(
        src, ss, g, s, E, fshift, o, os, oc);
  };
  auto cnt = [&](const int* idx, long long n, float* c) {
    k_count<<<gcap((n + 255) / 256), 256, 0, stream>>>(idx, n, c);
  };
  auto divi = [&](float* buf, int stride, int col, int dim, const float* c, int n) {
    k_div<<<gcap(((long long)n * dim + 255) / 256), 256, 0, stream>>>(buf, stride, col, dim, c, n);
  };
  auto gemm = [&](const float* A, int lda, const float* Wt, int ldw, const float* C0,
                  const float* bias, float* o, int N, int K, int Cout, int elu) {
    dim3 grid((N + 15) / 16), block(32 * (Cout / 16));
    switch (K) {
      case 32:  k_wmma_gemm<32><<<grid, block, 0, stream>>>(A, lda, Wt, ldw, C0, bias, o, N, Cout, elu); break;
      case 50:  k_wmma_gemm<50><<<grid, block, 0, stream>>>(A, lda, Wt, ldw, C0, bias, o, N, Cout, elu); break;
      case 64:  k_wmma_gemm<64><<<grid, block, 0, stream>>>(A, lda, Wt, ldw, C0, bias, o, N, Cout, elu); break;
      case 100: k_wmma_gemm<100><<<grid, block, 0, stream>>>(A, lda, Wt, ldw, C0, bias, o, N, Cout, elu); break;
      case 192: k_wmma_gemm<192><<<grid, block, 0, stream>>>(A, lda, Wt, ldw, C0, bias, o, N, Cout, elu); break;
      default: break;
    }
  };

  // ---- level 1: three GraphConvs on (N1, E1). lin(seg_sum(x)) == seg_sum(lin(x)) ----
  // conv1: 64 -> 32
  gemm(x, 64, c_wl[0], 64, nullptr, nullptr, XR, N1c, 64, 32, 0);
  zero(AG, (long long)N1c * 32);
  scat(XR, 32, ei1, ei1 + E1c, E1c, 32, AG, 32, 0);
  gemm(x, 64, c_wr[0], 64, AG, c_b[0], HB, N1c, 64, 32, 1);
  // conv2: 32 -> 64
  gemm(HB, 32, c_wl[1], 32, nullptr, nullptr, XR, N1c, 32, 64, 0);
  zero(AG, (long long)N1c * 64);
  scat(XR, 64, ei1, ei1 + E1c, E1c, 64, AG, 64, 0);
  gemm(HB, 32, c_wr[1], 32, AG, c_b[1], HA, N1c, 32, 64, 1);
  // conv3: 64 -> 64
  gemm(HA, 64, c_wl[2], 64, nullptr, nullptr, XR, N1c, 64, 64, 0);
  zero(AG, (long long)N1c * 64);
  scat(XR, 64, ei1, ei1 + E1c, E1c, 64, AG, 64, 0);
  gemm(HA, 64, c_wr[2], 64, AG, c_b[2], HB, N1c, 64, 64, 1);   // h -> HB (kept alive)

  // x_1 = seg_mean(h, batch) -> G[:, 0:64]
  zero(G, 256LL * 192);
  zero(CB, 256);
  scat(HB, 64, nullptr, bat1, N1c, 64, G, 192, 0);
  cnt(bat1, N1c, CB);
  divi(G, 192, 0, 64, CB, Bc);

  // ---- level 2: pool h to 2-tuples, two GraphConvs on (N2, E2), feat = [pooled|iso2] ----
  zero(P2, (long long)N2c * 64);
  zero(CN, N2c);
  scat(HB, 64, ai2, ai2 + A2c, A2c, 64, P2, 64, 0);
  cnt(ai2 + A2c, A2c, CN);
  divi(P2, 64, 0, 64, CN, N2c);
  // conv4: 114 -> 64, K split into pooled(64) + iso(50) via W column slices
  gemm(P2, 64, c_wl[3], 114, nullptr, nullptr, XR, N2c, 64, 64, 0);
  gemm(iso2, NI2c, c_wl[3] + 64, 114, XR, nullptr, XR, N2c, NI2c, 64, 0);
  zero(AG, (long long)N2c * 64);
  scat(XR, 64, ei2, ei2 + E2c, E2c, 64, AG, 64, 0);
  gemm(P2, 64, c_wr[3], 114, AG, nullptr, HC, N2c, 64, 64, 0);
  gemm(iso2, NI2c, c_wr[3] + 64, 114, HC, c_b[3], HC, N2c, NI2c, 64, 1);
  // conv5: 64 -> 64
  gemm(HC, 64, c_wl[4], 64, nullptr, nullptr, XR, N2c, 64, 64, 0);
  zero(AG, (long long)N2c * 64);
  scat(XR, 64, ei2, ei2 + E2c, E2c, 64, AG, 64, 0);
  gemm(HC, 64, c_wr[4], 64, AG, c_b[4], P2, N2c, 64, 64, 1);
  // x_2 -> G[:, 64:128]
  zero(CB, 256);
  scat(P2, 64, nullptr, bat2, N2c, 64, G, 192, 64);
  cnt(bat2, N2c, CB);
  divi(G, 192, 64, 64, CB, Bc);

  // ---- level 3: pool h to 3-tuples, two GraphConvs on (N3, E3), feat = [pooled|iso3] ----
  zero(P2, (long long)N3c * 64);
  zero(CN, N3c);
  scat(HB, 64, ai3, ai3 + A3c, A3c, 64, P2, 64, 0);
  cnt(ai3 + A3c, A3c, CN);
  divi(P2, 64, 0, 64, CN, N3c);
  // conv6: 164 -> 64, K split 64 + 100
  gemm(P2, 64, c_wl[5], 164, nullptr, nullptr, XR, N3c, 64, 64, 0);
  gemm(iso3, NI3c, c_wl[5] + 64, 164, XR, nullptr, XR, N3c, NI3c, 64, 0);
  zero(AG, (long long)N3c * 64);
  scat(XR, 64, ei3, ei3 + E3c, E3c, 64, AG, 64, 0);
  gemm(P2, 64, c_wr[5], 164, AG, nullptr, HC, N3c, 64, 64, 0);
  gemm(iso3, NI3c, c_wr[5] + 64, 164, HC, c_b[5], HC, N3c, NI3c, 64, 1);
  // conv7: 64 -> 64
  gemm(HC, 64, c_wl[6], 64, nullptr, nullptr, XR, N3c, 64, 64, 0);
  zero(AG, (long long)N3c * 64);
  scat(XR, 64, ei3, ei3 + E3c, E3c, 64, AG, 64, 0);
  gemm(HC, 64, c_wr[6], 64, AG, c_b[6], P2, N3c, 64, 64, 1);
  // x_3 -> G[:, 128:192]
  zero(CB, 256);
  scat(P2, 64, nullptr, bat3, N3c, 64, G, 192, 128);
  cnt(bat3, N3c, CB);
  divi(G, 192, 128, 64, CB, Bc);

  // ---- MLP head: fc1(192->64)+ELU, fc2(64->32)+ELU, fc3(32->11)+log_softmax ----
  float* T1 = HA;                // HA free at this point
  float* T2 = HA + 256LL * 64;
  gemm(G, 192, fw1, 192, nullptr, fb1, T1, Bc, 192, 64, 1);
  gemm(T1, 64, fw2, 64, nullptr, fb2, T2, Bc, 64, 32, 1);
  k_fc3_lsm<<<Bc, 32, 0, stream>>>(T2, fw3, fb3, (float*)d_out);
}